// GraphAttentionLayer_63513976373850
// MI455X (gfx1250) — compile-verified
//
#include <hip/hip_runtime.h>
#include <hip/hip_bf16.h>

#define N_NODES 8192
#define IN_FEAT 256
#define HID_F   256
#define ALPHA   0.2f

typedef float v2f __attribute__((ext_vector_type(2)));
typedef float v4f __attribute__((ext_vector_type(4)));
typedef float v8f __attribute__((ext_vector_type(8)));

// ---------------------------------------------------------------------------
// Kernel 1: fold the attention vectors through W once:
//   u1[k] = sum_n W[k][n] * a1[n],  u2[k] = sum_n W[k][n] * a2[n]
// Store as U (256 x 16 row-major): col0 = u1, col1 = u2, cols 2..15 = 0,
// ready to be consumed as the B operand of fp32 WMMA.
// ---------------------------------------------------------------------------
__global__ void gat_build_u(const float* __restrict__ W,
                            const float* __restrict__ a,
                            float* __restrict__ U) {
  int k = blockIdx.x * blockDim.x + threadIdx.x;
  if (k >= IN_FEAT) return;
  const float* wrow = W + k * HID_F;
  float u1 = 0.f, u2 = 0.f;
#pragma unroll 8
  for (int n = 0; n < HID_F; ++n) {
    float w = wrow[n];
    u1 = fmaf(w, a[n], u1);
    u2 = fmaf(w, a[HID_F + n], u2);
  }
  float* urow = U + k * 16;
  urow[0] = u1;
  urow[1] = u2;
#pragma unroll
  for (int c = 2; c < 16; ++c) urow[c] = 0.f;
}

// ---------------------------------------------------------------------------
// Kernel 2: S = src (8192x256) @ U (256x16) in fp32 WMMA (16x16x4).
// One wave per 16-row block: 64 accumulating WMMAs over K.
// A 16x4 layout: lane&15 = M; lanes>=16 supply K=2,3 (VGPR0/1 = K, K+1).
// B 4x16 layout: VGPRv -> K = v + 2*(lane>=16), N = lane&15.
// D 16x16 layout: VGPRv -> M = v + 8*(lane>=16), N = lane&15.
// Only N=0 (s1) and N=1 (s2) are written out.
// ---------------------------------------------------------------------------
__global__ void __launch_bounds__(256)
gat_scores_wmma(const float* __restrict__ src, const float* __restrict__ U,
                float* __restrict__ s1, float* __restrict__ s2) {
  int wave = (int)((blockIdx.x * blockDim.x + threadIdx.x) >> 5);
  int lane = threadIdx.x & 31;
  int col  = lane & 15;
  int half = lane >> 4;        // 0: lanes 0-15, 1: lanes 16-31
  int kOff = half * 2;

  int row = wave * 16 + col;   // A-matrix row handled by this lane
  const float* srcRow = src + row * IN_FEAT + kOff;
  const float* Ub     = U + kOff * 16 + col;

  v8f acc = {};
#pragma unroll 4
  for (int k0 = 0; k0 < IN_FEAT; k0 += 4) {
    v2f A, B;
    A.x = srcRow[k0];
    A.y = srcRow[k0 + 1];
    B.x = Ub[k0 * 16];        // K = k0 + kOff
    B.y = Ub[k0 * 16 + 16];   // K = k0 + kOff + 1
    acc = __builtin_amdgcn_wmma_f32_16x16x4_f32(
        /*neg_a=*/false, A, /*neg_b=*/false, B,
        /*c_mod=*/(short)0, acc, /*reuse_a=*/false, /*reuse_b=*/false);
  }

  if (col < 2) {
    float* dst = (col == 0) ? s1 : s2;
    int mBase = wave * 16 + half * 8;
#pragma unroll
    for (int v = 0; v < 8; ++v) dst[mBase + v] = acc[v];
  }
}

// ---------------------------------------------------------------------------
// Kernel 3 (bandwidth-critical): fused leakyrelu + bias + row softmax.
// One block per row; 256 threads x 8 float4 = full 32KB row in registers.
// Single streaming pass over bias (NT load), NT store of the output.
// ---------------------------------------------------------------------------
__global__ void __launch_bounds__(256)
gat_softmax(const float* __restrict__ bias,
            const float* __restrict__ s1, const float* __restrict__ s2,
            float* __restrict__ out) {
  const int row = blockIdx.x;
  const int tid = threadIdx.x;
  const float si = s1[row];

  const v4f* brow = (const v4f*)(bias + (size_t)row * N_NODES);
  v4f*       orow = (v4f*)(out + (size_t)row * N_NODES);
  const v4f* s2v  = (const v4f*)s2;

  v4f vals[8];
  float lmax = -INFINITY;
#pragma unroll
  for (int t = 0; t < 8; ++t) {
    int j4 = t * 256 + tid;                       // 2048 float4 per row
    v4f b  = __builtin_nontemporal_load(&brow[j4]);
    v4f sv = s2v[j4];
    v4f e;
#pragma unroll
    for (int c = 0; c < 4; ++c) {
      float x = si + sv[c];
      x = (x > 0.f) ? x : (ALPHA * x);            // leaky relu
      x += b[c];
      e[c] = x;
      lmax = fmaxf(lmax, x);
    }
    vals[t] = e;
  }

  __shared__ float red[256];
  red[tid] = lmax;
  __syncthreads();
  for (int s = 128; s > 0; s >>= 1) {
    if (tid < s) red[tid] = fmaxf(red[tid], red[tid + s]);
    __syncthreads();
  }
  const float gmax = red[0];
  __syncthreads();

  float lsum = 0.f;
#pragma unroll
  for (int t = 0; t < 8; ++t) {
#pragma unroll
    for (int c = 0; c < 4; ++c) {
      float x = __expf(vals[t][c] - gmax);
      vals[t][c] = x;
      lsum += x;
    }
  }
  red[tid] = lsum;
  __syncthreads();
  for (int s = 128; s > 0; s >>= 1) {
    if (tid < s) red[tid] += red[tid + s];
    __syncthreads();
  }
  const float inv = 1.0f / red[0];

#pragma unroll
  for (int t = 0; t < 8; ++t) {
    int j4 = t * 256 + tid;
    v4f o = vals[t] * inv;
    __builtin_nontemporal_store(o, &orow[j4]);
  }
}

// ---------------------------------------------------------------------------
extern "C" void kernel_launch(void* const* d_in, const int* in_sizes, int n_in,
                              void* d_out, int out_size, void* d_ws, size_t ws_size,
                              hipStream_t stream) {
  const float* src  = (const float*)d_in[0];   // (8192, 256)
  const float* bias = (const float*)d_in[1];   // (8192, 8192)
  const float* W    = (const float*)d_in[2];   // (256, 256)
  const float* a    = (const float*)d_in[3];   // (512, 1)
  float* out = (float*)d_out;                  // (8192, 8192)

  float* U  = (float*)d_ws;                    // 256*16 floats
  float* s1 = U + IN_FEAT * 16;                // 8192 floats
  float* s2 = s1 + N_NODES;                    // 8192 floats

  gat_build_u<<<1, 256, 0, stream>>>(W, a, U);
  // 512 waves total (8192/16 row-blocks) => 64 blocks x 256 threads (8 waves)
  gat_scores_wmma<<<64, 256, 0, stream>>>(src, U, s1, s2);
  gat_softmax<<<N_NODES, 256, 0, stream>>>(bias, s1, s2, out);
}